// GraphSageLayer_83932250898900
// MI455X (gfx1250) — compile-verified
//
#include <hip/hip_runtime.h>
#include <hip/hip_bf16.h>

// ---------------------------------------------------------------------------
// GraphSAGE layer for MI455X (gfx1250, wave32, WMMA)
//   neigh = segment_mean(h[src], dst)
//   rst   = relu(h @ W_self + neigh @ W_neigh + b)
//   out   = h + BN_train(rst) * gamma + beta   (residual)
// ---------------------------------------------------------------------------

#define NNODES 100000
#define NEDGES 3200000
#define DFEAT  128
#define EPS    1e-5f

typedef __attribute__((ext_vector_type(2))) float v2f;
typedef __attribute__((ext_vector_type(8))) float v8f;

// workspace layout (floats):
//   [0, N*D)                : neigh_sum, later overwritten with rst
//   [N*D, N*D+N)            : deg
//   [N*D+N, +128)           : colsum
//   [.., +128)              : colsumsq
//   [.., +128)              : scale
//   [.., +128)              : shift
#define WS_RST      0
#define WS_DEG      (NNODES * DFEAT)
#define WS_COLSUM   (WS_DEG + NNODES)
#define WS_COLSQ    (WS_COLSUM + DFEAT)
#define WS_SCALE    (WS_COLSQ + DFEAT)
#define WS_SHIFT    (WS_SCALE + DFEAT)

// ---------------------------------------------------------------------------
// Stage 2: edge scatter.  One wave (32 lanes) per edge; each lane owns 4
// consecutive features (float4 read of h[src], 4 float atomics into neigh_sum).
// ---------------------------------------------------------------------------
__global__ void sage_edge_scatter(const float* __restrict__ h,
                                  const int* __restrict__ src,
                                  const int* __restrict__ dst,
                                  float* __restrict__ ws) {
  int t = blockIdx.x * blockDim.x + threadIdx.x;
  int e = t >> 5;
  if (e >= NEDGES) return;
  int lane = t & 31;
  int s = src[e];
  int d = dst[e];
  const float4* h4 = (const float4*)h;
  float4 v = h4[s * (DFEAT / 4) + lane];
  float* acc = ws + WS_RST + (size_t)d * DFEAT + lane * 4;
  unsafeAtomicAdd(acc + 0, v.x);
  unsafeAtomicAdd(acc + 1, v.y);
  unsafeAtomicAdd(acc + 2, v.z);
  unsafeAtomicAdd(acc + 3, v.w);
  if (lane == 0) unsafeAtomicAdd(ws + WS_DEG + d, 1.0f);
}

// ---------------------------------------------------------------------------
// Stage 3: fused dual-GEMM + bias + ReLU + BN-stats, using v_wmma_f32_16x16x4_f32.
// Block = 256 threads = 8 waves. Block owns 16 rows; wave w owns column tile
// [16w, 16w+16).  h tile and neigh tile staged in LDS (16 KB).
// rst written back over the neigh_sum rows (safe: per-block-private rows,
// staged before the barrier).
// ---------------------------------------------------------------------------
__global__ void __launch_bounds__(256)
sage_gemm_wmma(const float* __restrict__ h,
               const float* __restrict__ Wself,
               const float* __restrict__ Wneigh,
               const float* __restrict__ bvec,
               float* __restrict__ ws) {
  __shared__ float sH[16][DFEAT];
  __shared__ float sN[16][DFEAT];

  const int r0 = blockIdx.x * 16;            // first row of this block
  const int tid = threadIdx.x;

  // stage A tiles: 2048 elements each, 8 per thread, coalesced
  const float* nsum = ws + WS_RST;
  const float* deg  = ws + WS_DEG;
#pragma unroll
  for (int j = 0; j < 8; ++j) {
    int idx = tid + j * 256;                 // 0..2047
    int row = idx >> 7;
    int ck  = idx & (DFEAT - 1);
    sH[row][ck] = h[(size_t)(r0 + row) * DFEAT + ck];
    float dg = deg[r0 + row];
    float inv = 1.0f / fmaxf(dg, 1.0f);
    sN[row][ck] = nsum[(size_t)(r0 + row) * DFEAT + ck] * inv;
  }
  __syncthreads();

  const int wave = tid >> 5;                 // 0..7 -> column tile
  const int lane = tid & 31;
  const int half = lane >> 4;                // 0/1
  const int m    = lane & 15;                // A-row / C-col index within tile
  const int col  = wave * 16 + (lane & 15);  // global output column

  v8f c = {};
#pragma unroll 4
  for (int kk = 0; kk < DFEAT; kk += 4) {
    int ka = kk + 2 * half;
    // A fragments (16x4 f32): lane (m, half) holds K = ka, ka+1
    v2f aH = *(const v2f*)&sH[m][ka];
    v2f aN = *(const v2f*)&sN[m][ka];
    // B fragments (4x16 f32): VGPR0 = row ka, VGPR1 = row ka+1, col striped
    v2f bS, bN;
    bS.x = Wself [(size_t)ka * DFEAT + col];
    bS.y = Wself [(size_t)(ka + 1) * DFEAT + col];
    bN.x = Wneigh[(size_t)ka * DFEAT + col];
    bN.y = Wneigh[(size_t)(ka + 1) * DFEAT + col];
    c = __builtin_amdgcn_wmma_f32_16x16x4_f32(false, aH, false, bS,
                                              (short)0, c, false, false);
    c = __builtin_amdgcn_wmma_f32_16x16x4_f32(false, aN, false, bN,
                                              (short)0, c, false, false);
  }

  // epilogue: bias + relu, write rst, accumulate BN column stats
  float bias = bvec[col];
  float s = 0.0f, s2 = 0.0f;
  float* rst = ws + WS_RST;
#pragma unroll
  for (int r = 0; r < 8; ++r) {
    float v = c[r] + bias;
    v = fmaxf(v, 0.0f);
    int row = r0 + r + 8 * half;             // C layout: VGPR r -> M = r + 8*half
    rst[(size_t)row * DFEAT + col] = v;
    s += v;
    s2 += v * v;
  }
  unsafeAtomicAdd(ws + WS_COLSUM + col, s);
  unsafeAtomicAdd(ws + WS_COLSQ + col, s2);
}

// ---------------------------------------------------------------------------
// Stage 4: fold column stats into affine scale/shift
// ---------------------------------------------------------------------------
__global__ void sage_bn_finalize(const float* __restrict__ gamma,
                                 const float* __restrict__ beta,
                                 float* __restrict__ ws) {
  int c = threadIdx.x;
  if (c >= DFEAT) return;
  float invN = 1.0f / (float)NNODES;
  float mean = ws[WS_COLSUM + c] * invN;
  float var  = ws[WS_COLSQ + c] * invN - mean * mean;
  float inv  = rsqrtf(var + EPS);
  float sc   = inv * gamma[c];
  ws[WS_SCALE + c] = sc;
  ws[WS_SHIFT + c] = beta[c] - mean * sc;
}

// ---------------------------------------------------------------------------
// Stage 5: out = h + rst*scale + shift  (float4 vectorized)
// ---------------------------------------------------------------------------
__global__ void sage_residual(const float* __restrict__ h,
                              const float* __restrict__ ws,
                              float* __restrict__ out) {
  int t = blockIdx.x * blockDim.x + threadIdx.x;
  const int total4 = NNODES * DFEAT / 4;
  if (t >= total4) return;
  int cb = (t * 4) & (DFEAT - 1);            // column of first component
  const float4 hv = ((const float4*)h)[t];
  const float4 rv = ((const float4*)(ws + WS_RST))[t];
  const float* sc = ws + WS_SCALE;
  const float* sh = ws + WS_SHIFT;
  float4 o;
  o.x = fmaf(rv.x, sc[cb + 0], sh[cb + 0]) + hv.x;
  o.y = fmaf(rv.y, sc[cb + 1], sh[cb + 1]) + hv.y;
  o.z = fmaf(rv.z, sc[cb + 2], sh[cb + 2]) + hv.z;
  o.w = fmaf(rv.w, sc[cb + 3], sh[cb + 3]) + hv.w;
  ((float4*)out)[t] = o;
}

// ---------------------------------------------------------------------------
extern "C" void kernel_launch(void* const* d_in, const int* in_sizes, int n_in,
                              void* d_out, int out_size, void* d_ws, size_t ws_size,
                              hipStream_t stream) {
  const float* h      = (const float*)d_in[0];
  const int*   src    = (const int*)d_in[1];
  const int*   dst    = (const int*)d_in[2];
  const float* Wself  = (const float*)d_in[3];
  const float* Wneigh = (const float*)d_in[4];
  const float* bvec   = (const float*)d_in[5];
  const float* gamma  = (const float*)d_in[6];
  const float* beta   = (const float*)d_in[7];
  float* out = (float*)d_out;
  float* ws  = (float*)d_ws;

  // zero accumulators: neigh_sum + deg + colsum + colsumsq
  size_t zero_bytes = (size_t)(NNODES * DFEAT + NNODES + 2 * DFEAT) * sizeof(float);
  hipMemsetAsync(d_ws, 0, zero_bytes, stream);

  // edge scatter: one wave per edge
  {
    long long threads = (long long)NEDGES * 32;
    int blocks = (int)((threads + 255) / 256);
    sage_edge_scatter<<<blocks, 256, 0, stream>>>(h, src, dst, ws);
  }

  // fused WMMA GEMM (+bias, relu, BN stats)
  sage_gemm_wmma<<<NNODES / 16, 256, 0, stream>>>(h, Wself, Wneigh, bvec, ws);

  // BN finalize
  sage_bn_finalize<<<1, DFEAT, 0, stream>>>(gamma, beta, ws);

  // residual + affine
  {
    int total4 = NNODES * DFEAT / 4;
    sage_residual<<<(total4 + 255) / 256, 256, 0, stream>>>(h, ws, out);
  }
}